// RatingPredictionGNN_16750372455064
// MI455X (gfx1250) — compile-verified
//
#include <hip/hip_runtime.h>

#define NUM_USERS 100000
#define NUM_BOOKS 50000
#define N_NODES   150000
#define LATENT    64
#define N_EDGES   2400000

typedef __attribute__((ext_vector_type(2))) float v2f;
typedef __attribute__((ext_vector_type(8))) float v8f;

// ---------------- utility kernels ----------------

__global__ void fill_zero_kernel(float* __restrict__ p, int n) {
    int i = blockIdx.x * blockDim.x + threadIdx.x;
    if (i < n) p[i] = 0.0f;
}

__global__ void degree_kernel(const int* __restrict__ to, float* __restrict__ deg) {
    int e = blockIdx.x * blockDim.x + threadIdx.x;
    if (e < N_EDGES) {
        unsafeAtomicAdd(&deg[to[e]], 1.0f);
    }
}

__global__ void dinv_kernel(const float* __restrict__ deg, float* __restrict__ dinv) {
    int i = blockIdx.x * blockDim.x + threadIdx.x;
    if (i < N_NODES) {
        float d = deg[i];
        dinv[i] = (d > 0.0f) ? (1.0f / sqrtf(d)) : 0.0f;
    }
}

// Edge-parallel gather/scale/scatter. 16 lanes per edge, float4 per lane.
// src rows are L2-resident (38.4MB table vs 192MB L2).
__global__ void propagate_kernel(const int* __restrict__ frm, const int* __restrict__ to,
                                 const float* __restrict__ dinv,
                                 const float* __restrict__ src, float* __restrict__ dst) {
    long long idx = (long long)blockIdx.x * blockDim.x + threadIdx.x;
    int e = (int)(idx >> 4);
    int q = (int)(idx & 15);
    if (e >= N_EDGES) return;
    int f = frm[e];
    int t = to[e];
    float w = dinv[f] * dinv[t];
    const float4 v = *(const float4*)(src + (size_t)f * LATENT + q * 4);
    float* d = dst + (size_t)t * LATENT + q * 4;
    unsafeAtomicAdd(d + 0, w * v.x);
    unsafeAtomicAdd(d + 1, w * v.y);
    unsafeAtomicAdd(d + 2, w * v.z);
    unsafeAtomicAdd(d + 3, w * v.w);
}

__global__ void acc_kernel(float* __restrict__ acc, const float* __restrict__ x, int n4) {
    int i = blockIdx.x * blockDim.x + threadIdx.x;
    if (i < n4) {
        float4 a = ((const float4*)acc)[i];
        float4 b = ((const float4*)x)[i];
        a.x += b.x; a.y += b.y; a.z += b.z; a.w += b.w;
        ((float4*)acc)[i] = a;
    }
}

__global__ void scale_kernel(float* __restrict__ p, float s, int n4) {
    int i = blockIdx.x * blockDim.x + threadIdx.x;
    if (i < n4) {
        float4 a = ((const float4*)p)[i];
        a.x *= s; a.y *= s; a.z *= s; a.w *= s;
        ((float4*)p)[i] = a;
    }
}

// ---------------- WMMA projection kernels ----------------
// One wave computes a 16x64 output tile: 4 accumulators of 16x16 (v8f each),
// K advanced 4 at a time via V_WMMA_F32_16X16X4_F32.
//
// f32 WMMA lane layouts (ISA 7.12.2):
//   A 16x4 : lane(half,m) -> a[j] = A[m][k0 + 2*half + j]
//   B 4x16 : lane(half,m) -> b[j] = B[k0 + 2*half + j][n0 + m]
//   D 16x16: lane(half,m), vgpr v -> D[v + 8*half][n0 + m]

#define WMMA_STEP4(Wp, k, a, c0, c1, c2, c3)                                            \
    do {                                                                                 \
        const float* _w0 = (Wp) + (size_t)(k) * LATENT + m;                              \
        const float* _w1 = (Wp) + (size_t)((k) + 1) * LATENT + m;                        \
        v2f _b;                                                                          \
        _b.x = _w0[0];  _b.y = _w1[0];                                                   \
        c0 = __builtin_amdgcn_wmma_f32_16x16x4_f32(false, a, false, _b, (short)0, c0, false, false); \
        _b.x = _w0[16]; _b.y = _w1[16];                                                  \
        c1 = __builtin_amdgcn_wmma_f32_16x16x4_f32(false, a, false, _b, (short)0, c1, false, false); \
        _b.x = _w0[32]; _b.y = _w1[32];                                                  \
        c2 = __builtin_amdgcn_wmma_f32_16x16x4_f32(false, a, false, _b, (short)0, c2, false, false); \
        _b.x = _w0[48]; _b.y = _w1[48];                                                  \
        c3 = __builtin_amdgcn_wmma_f32_16x16x4_f32(false, a, false, _b, (short)0, c3, false, false); \
    } while (0)

__global__ __launch_bounds__(128)
void emb0_user_kernel(const float* __restrict__ uf,   // [Nu,16]
                      const float* __restrict__ Wu,   // [16,64]
                      const float* __restrict__ bu,   // [64]
                      const float* __restrict__ emb,  // [N,64]
                      float* __restrict__ emb0,       // [N,64]
                      float* __restrict__ acc) {      // [N,64]
    const int tile = blockIdx.x * 4 + (threadIdx.x >> 5);
    if (tile >= NUM_USERS / 16) return;       // uniform per wave -> EXEC stays all-ones
    const int lane = threadIdx.x & 31;
    const int half = lane >> 4;
    const int m = lane & 15;
    const int row0 = tile * 16;

    v8f c0 = {}, c1 = {}, c2 = {}, c3 = {};
    const float* arow = uf + (size_t)(row0 + m) * 16;
#pragma unroll
    for (int kk = 0; kk < 4; ++kk) {          // K = 16
        const int k = kk * 4 + 2 * half;
        v2f a; a.x = arow[k]; a.y = arow[k + 1];
        WMMA_STEP4(Wu, k, a, c0, c1, c2, c3);
    }
#pragma unroll
    for (int v = 0; v < 8; ++v) {
        const int R = row0 + v + 8 * half;
        const size_t base = (size_t)R * LATENT;
        float r0 = emb[base +  0 + m] + c0[v] + bu[ 0 + m];
        float r1 = emb[base + 16 + m] + c1[v] + bu[16 + m];
        float r2 = emb[base + 32 + m] + c2[v] + bu[32 + m];
        float r3 = emb[base + 48 + m] + c3[v] + bu[48 + m];
        emb0[base +  0 + m] = r0;  acc[base +  0 + m] = r0;
        emb0[base + 16 + m] = r1;  acc[base + 16 + m] = r1;
        emb0[base + 32 + m] = r2;  acc[base + 32 + m] = r2;
        emb0[base + 48 + m] = r3;  acc[base + 48 + m] = r3;
    }
}

__global__ __launch_bounds__(128)
void emb0_book_kernel(const float* __restrict__ bnf,  // [Nb,8]
                      const float* __restrict__ bgf,  // [Nb,32]
                      const float* __restrict__ Wn,   // [8,64]
                      const float* __restrict__ bn,   // [64]
                      const float* __restrict__ Wg,   // [32,64]
                      const float* __restrict__ bg,   // [64]
                      const float* __restrict__ emb,  // [N,64]
                      float* __restrict__ emb0,
                      float* __restrict__ acc) {
    const int tile = blockIdx.x * 4 + (threadIdx.x >> 5);
    if (tile >= NUM_BOOKS / 16) return;
    const int lane = threadIdx.x & 31;
    const int half = lane >> 4;
    const int m = lane & 15;
    const int row0 = tile * 16;

    v8f c0 = {}, c1 = {}, c2 = {}, c3 = {};
    const float* arowN = bnf + (size_t)(row0 + m) * 8;
    const float* arowG = bgf + (size_t)(row0 + m) * 32;
#pragma unroll
    for (int kk = 0; kk < 2; ++kk) {          // numerical features, K = 8
        const int k = kk * 4 + 2 * half;
        v2f a; a.x = arowN[k]; a.y = arowN[k + 1];
        WMMA_STEP4(Wn, k, a, c0, c1, c2, c3);
    }
#pragma unroll
    for (int kk = 0; kk < 8; ++kk) {          // genre features, K = 32
        const int k = kk * 4 + 2 * half;
        v2f a; a.x = arowG[k]; a.y = arowG[k + 1];
        WMMA_STEP4(Wg, k, a, c0, c1, c2, c3);
    }
#pragma unroll
    for (int v = 0; v < 8; ++v) {
        const int R = NUM_USERS + row0 + v + 8 * half;
        const size_t base = (size_t)R * LATENT;
        float r0 = emb[base +  0 + m] + c0[v] + bn[ 0 + m] + bg[ 0 + m];
        float r1 = emb[base + 16 + m] + c1[v] + bn[16 + m] + bg[16 + m];
        float r2 = emb[base + 32 + m] + c2[v] + bn[32 + m] + bg[32 + m];
        float r3 = emb[base + 48 + m] + c3[v] + bn[48 + m] + bg[48 + m];
        emb0[base +  0 + m] = r0;  acc[base +  0 + m] = r0;
        emb0[base + 16 + m] = r1;  acc[base + 16 + m] = r1;
        emb0[base + 32 + m] = r2;  acc[base + 32 + m] = r2;
        emb0[base + 48 + m] = r3;  acc[base + 48 + m] = r3;
    }
}

// ---------------- launch ----------------

extern "C" void kernel_launch(void* const* d_in, const int* in_sizes, int n_in,
                              void* d_out, int out_size, void* d_ws, size_t ws_size,
                              hipStream_t stream) {
    const int*   ei   = (const int*)  d_in[0];   // [2, E]
    const float* emb  = (const float*)d_in[1];   // [N, 64]
    const float* uf   = (const float*)d_in[2];   // [Nu, 16]
    const float* bnf  = (const float*)d_in[3];   // [Nb, 8]
    const float* bgf  = (const float*)d_in[4];   // [Nb, 32]
    const float* Wu   = (const float*)d_in[5];   // [16, 64]
    const float* bu   = (const float*)d_in[6];   // [64]
    const float* Wn   = (const float*)d_in[7];   // [8, 64]
    const float* bn   = (const float*)d_in[8];   // [64]
    const float* Wg   = (const float*)d_in[9];   // [32, 64]
    const float* bg   = (const float*)d_in[10];  // [64]

    float* emb0 = (float*)d_out;                            // first output
    float* acc  = emb0 + (size_t)N_NODES * LATENT;          // second output (layer mean)

    float* deg  = (float*)d_ws;
    float* dinv = deg + N_NODES;
    float* embA = dinv + N_NODES;
    float* embB = embA + (size_t)N_NODES * LATENT;

    const int* frm = ei;
    const int* to  = ei + N_EDGES;

    const int NV  = N_NODES * LATENT;           // 9.6M
    const int NV4 = NV / 4;
    const int zgrid = (NV + 255) / 256;
    const int agrid = (NV4 + 255) / 256;
    const long long pthreads = (long long)N_EDGES * 16;
    const int pgrid = (int)((pthreads + 255) / 256);

    // degrees -> deg^-1/2
    fill_zero_kernel<<<(N_NODES + 255) / 256, 256, 0, stream>>>(deg, N_NODES);
    degree_kernel<<<(N_EDGES + 255) / 256, 256, 0, stream>>>(to, deg);
    dinv_kernel<<<(N_NODES + 255) / 256, 256, 0, stream>>>(deg, dinv);

    // emb0 = embedding + projections (WMMA); also initializes acc = emb0
    emb0_user_kernel<<<(NUM_USERS / 16 + 3) / 4, 128, 0, stream>>>(uf, Wu, bu, emb, emb0, acc);
    emb0_book_kernel<<<(NUM_BOOKS / 16 + 3) / 4, 128, 0, stream>>>(bnf, bgf, Wn, bn, Wg, bg,
                                                                    emb, emb0, acc);

    // layer 1: emb0 -> embA
    fill_zero_kernel<<<zgrid, 256, 0, stream>>>(embA, NV);
    propagate_kernel<<<pgrid, 256, 0, stream>>>(frm, to, dinv, emb0, embA);
    acc_kernel<<<agrid, 256, 0, stream>>>(acc, embA, NV4);

    // layer 2: embA -> embB
    fill_zero_kernel<<<zgrid, 256, 0, stream>>>(embB, NV);
    propagate_kernel<<<pgrid, 256, 0, stream>>>(frm, to, dinv, embA, embB);
    acc_kernel<<<agrid, 256, 0, stream>>>(acc, embB, NV4);

    // layer 3: embB -> embA
    fill_zero_kernel<<<zgrid, 256, 0, stream>>>(embA, NV);
    propagate_kernel<<<pgrid, 256, 0, stream>>>(frm, to, dinv, embB, embA);
    acc_kernel<<<agrid, 256, 0, stream>>>(acc, embA, NV4);

    // out = acc / (NUM_LAYERS + 1)
    scale_kernel<<<agrid, 256, 0, stream>>>(acc, 0.25f, NV4);
}